// SPGG_Qlearning_14242111553552
// MI455X (gfx1250) — compile-verified
//
#include <hip/hip_runtime.h>
#include <hip/hip_bf16.h>
#include <stdint.h>

// ---- constants from the reference ----
#define LSZ     2048
#define LMASK   (LSZ - 1)
#define ALPHA   0.8f
#define GAMMA   0.8f
#define EPSV    (-1.0f)
#define RVAL    3.8f
#define KFERMI  0.5f
#define NEGBIG  1.0e9f

typedef float        __attribute__((ext_vector_type(4))) f32x4;
typedef float        __attribute__((ext_vector_type(2))) f32x2;
typedef unsigned int __attribute__((ext_vector_type(4))) u32x4;
typedef int          __attribute__((ext_vector_type(4))) i32x4;
typedef int          __attribute__((ext_vector_type(8))) i32x8;

// ---- probe CDNA5 builtins and report via stderr (#warning) ----
#if defined(__has_builtin)
#if __has_builtin(__builtin_amdgcn_global_load_async_to_lds_b32)
#define USE_ASYNC_LDS 1
#warning "CDNA5 probe: __builtin_amdgcn_global_load_async_to_lds_b32 AVAILABLE"
#else
#warning "CDNA5 probe: __builtin_amdgcn_global_load_async_to_lds_b32 NOT available"
#endif
#if __has_builtin(__builtin_amdgcn_tensor_load_to_lds)
#define USE_TDM 1
#warning "CDNA5 probe: __builtin_amdgcn_tensor_load_to_lds AVAILABLE"
#else
#warning "CDNA5 probe: __builtin_amdgcn_tensor_load_to_lds NOT available"
#endif
#if __clang_major__ >= 23
#warning "CDNA5 probe: clang >= 23 (6-arg TDM builtin)"
#else
#warning "CDNA5 probe: clang <= 22 (5-arg TDM builtin)"
#endif
#endif
#ifndef USE_ASYNC_LDS
#define USE_ASYNC_LDS 0
#endif
#ifndef USE_TDM
#define USE_TDM 0
#endif

// tile geometry: 32x32 outputs, halo 3 on type, halo 2 on coop, halo 1 on p2/B
#define TILE 32
#define THD  38   // type tile rows (TILE + 2*3)
#define TSTR 40   // type tile row stride (TDM loads 40 cols per row: 160B contiguous)
#define CHD  36   // coop tile dim (TILE + 2*2)
#define CSTR 38
#define PHD  34   // p2/B tile dim (TILE + 2*1)
#define PSTR 36

__global__ __launch_bounds__(256)
void spgg_fused_kernel(const int*   __restrict__ type_in,
                       const float* __restrict__ Q_in,
                       const float* __restrict__ tie_in,
                       const int*   __restrict__ rtype_in,
                       const float* __restrict__ eps_in,
                       const int*   __restrict__ dir_in,
                       const float* __restrict__ lp_in,
                       float* __restrict__ out_type,
                       float* __restrict__ out_Q,
                       float* __restrict__ out_profit)
{
    __shared__ int   s_type[THD * TSTR];  // 6080 B (38 rows x 40 cols)
    __shared__ float s_coop[CHD * CSTR];
    __shared__ float s_p2  [PHD * PSTR];
    __shared__ int   s_B   [PHD * PSTR];

    const int tid = threadIdx.x;
    const int ox  = blockIdx.x * TILE;
    const int oy  = blockIdx.y * TILE;

    // interior blocks: the 38x40 halo tile never wraps -> eligible for TDM DMA
    const bool interior = (oy >= 3) && (oy + THD - 3 <= LSZ) &&
                          (ox >= 3) && (ox + TSTR - 3 <= LSZ);

    // ---- phase 0: prefetch the Q cachelines this tile will consume ----
    for (int i = tid; i < PHD * PHD; i += 256) {
        int s = i / PHD, t = i - s * PHD;
        if ((t & 7) == 0) {
            int gy = (oy + s - 1) & LMASK;
            int gx = (ox + t - 1) & LMASK;
            __builtin_prefetch(Q_in + (size_t)(gy * LSZ + gx) * 4, 0, 0);
        }
    }

    // ---- phase 1: stage type tile (+3 halo) into LDS ----
#if USE_TDM
    if (interior) {
        if (tid < 32) {  // one wave issues the tensor DMA (EXEC-independent op)
            unsigned long long ga =
                (unsigned long long)(uintptr_t)(type_in +
                    (size_t)(oy - 3) * LSZ + (size_t)(ox - 3));
            unsigned lds_base =
                (unsigned)(size_t)(__attribute__((address_space(3))) int*)s_type;

            // D# group0: count=1 | lds_addr | global_addr[56:0] | type=2
            u32x4 g0;
            g0.x = 1u;
            g0.y = lds_base;
            g0.z = (unsigned)(ga & 0xffffffffull);
            g0.w = (unsigned)((ga >> 32) & 0x01ffffffull) | 0x80000000u; // type=2

            // D# group1: data_size=4B, tile 40x38, row stride 2048, huge tensor dims
            i32x8 g1;
            g1[0] = 0x00020000;              // workgroup_mask=0, data_size=2 (4B)
            g1[1] = 0;                        // no atomic barrier; tensor_dim0 lo16=0
            g1[2] = 0x4000;                   // tensor_dim0 hi16 = 0x4000 (1<<30)
            g1[3] = (TSTR << 16) | 0x4000;    // tile_dim0=40 | tensor_dim1 hi16
            g1[4] = THD;                      // tile_dim1=38, tile_dim2=0 (2D)
            g1[5] = LSZ;                      // tensor_dim0_stride = 2048 elements
            g1[6] = 0;
            g1[7] = 0;

            i32x4 g2 = (i32x4)0;              // dims 2/3 unused
            i32x4 g3 = (i32x4)0;              // dim 4 unused
#if __clang_major__ >= 23
            __builtin_amdgcn_tensor_load_to_lds(g0, g1, g2, g3, (i32x8)0, 0);
#else
            __builtin_amdgcn_tensor_load_to_lds(g0, g1, g2, g3, 0);
#endif
            __builtin_amdgcn_s_wait_tensorcnt(0);
        }
    } else
#endif
    {
        for (int i = tid; i < THD * THD; i += 256) {
            int ty = i / THD, tx = i - ty * THD;
            int gy = (oy + ty - 3) & LMASK;
            int gx = (ox + tx - 3) & LMASK;
#if USE_ASYNC_LDS
            __builtin_amdgcn_global_load_async_to_lds_b32(
                (__attribute__((address_space(1))) int*)(type_in + gy * LSZ + gx),
                (__attribute__((address_space(3))) int*)&s_type[ty * TSTR + tx],
                0, 0);
#else
            s_type[ty * TSTR + tx] = type_in[gy * LSZ + gx];
#endif
        }
#if USE_ASYNC_LDS
#if __has_builtin(__builtin_amdgcn_s_wait_asynccnt)
        __builtin_amdgcn_s_wait_asynccnt(0);
#else
        asm volatile("s_wait_asynccnt 0" ::: "memory");
#endif
#endif
    }
    __syncthreads();

    // ---- phase 2: coop = neighbor5(type==1) on the +2-halo region ----
    for (int i = tid; i < CHD * CHD; i += 256) {
        int u = i / CHD, v = i - u * CHD;
        int b = (u + 1) * TSTR + (v + 1);
        int cc = (s_type[b] == 1) + (s_type[b - TSTR] == 1) + (s_type[b + TSTR] == 1)
               + (s_type[b - 1] == 1) + (s_type[b + 1] == 1);
        s_coop[u * CSTR + v] = (float)cc;
    }
    __syncthreads();

    // ---- phase 3+4: profit, epsilon-greedy, Q update; p2/B on +1 halo ----
    for (int i = tid; i < PHD * PHD; i += 256) {
        int s = i / PHD, t = i - s * PHD;

        int cb = (s + 1) * CSTR + (t + 1);
        float n2 = s_coop[cb] + s_coop[cb - CSTR] + s_coop[cb + CSTR]
                 + s_coop[cb - 1] + s_coop[cb + 1];
        int tv = s_type[(s + 2) * TSTR + (t + 2)];
        float cf = (tv == 1) ? 1.0f : 0.0f;
        float df = (tv == 0) ? 1.0f : 0.0f;
        float c5 = n2 * (RVAL / 5.0f) - 5.0f;
        float d5 = n2 * (RVAL / 5.0f);
        float profit = c5 * cf + d5 * df;

        int gy = (oy + s - 1) & LMASK;
        int gx = (ox + t - 1) & LMASK;
        int g  = gy * LSZ + gx;

        f32x4 q  = __builtin_nontemporal_load((const f32x4*)Q_in + g);
        f32x2 tr = __builtin_nontemporal_load((const f32x2*)tie_in + g);
        float er = __builtin_nontemporal_load(eps_in + g);
        int   rt = __builtin_nontemporal_load(rtype_in + g);

        int A = tv;                       // action at t (0/1)
        float qa0 = A ? q.z : q.x;
        float qa1 = A ? q.w : q.y;
        float mv  = fmaxf(qa0, qa1);
        float sc0 = ((qa0 == mv) ? 1.0f : -NEGBIG) + tr.x;
        float sc1 = ((qa1 == mv) ? 1.0f : -NEGBIG) + tr.y;
        int greedy = (sc1 > sc0) ? 1 : 0;
        int B = (er >= EPSV) ? greedy : rt;

        float qb0 = B ? q.z : q.x;
        float qb1 = B ? q.w : q.y;
        float max_next = fmaxf(qb0, qb1);
        float q_old = A ? (B ? q.w : q.z) : (B ? q.y : q.x);
        float q_new = q_old + ALPHA * (profit + GAMMA * max_next - q_old);

        s_p2[s * PSTR + t] = q_new;
        s_B [s * PSTR + t] = B;

        if (s >= 1 && s < PHD - 1 && t >= 1 && t < PHD - 1) {
            f32x4 qo = q;
            if (A == 0) { if (B == 0) qo.x = q_new; else qo.y = q_new; }
            else        { if (B == 0) qo.z = q_new; else qo.w = q_new; }
            __builtin_nontemporal_store(qo, (f32x4*)out_Q + g);
            __builtin_nontemporal_store(profit, out_profit + g);
        }
    }
    __syncthreads();

    // ---- phase 5: Fermi imitation from the direction-selected neighbor ----
    for (int i = tid; i < TILE * TILE; i += 256) {
        int ly = i >> 5, lx = i & 31;
        int s = ly + 1, t = lx + 1;
        int gy = (oy + ly) & LMASK;
        int gx = (ox + lx) & LMASK;
        int g  = gy * LSZ + gx;

        int dir = __builtin_nontemporal_load(dir_in + g) & 3;
        int dy = (dir == 2) ? -1 : ((dir == 3) ? 1 : 0);
        int dx = (dir == 0) ? -1 : ((dir == 1) ? 1 : 0);

        float p2c = s_p2[s * PSTR + t];
        float p2n = s_p2[(s + dy) * PSTR + (t + dx)];
        int   bn  = s_B [(s + dy) * PSTR + (t + dx)];
        float w = 1.0f / (1.0f + expf((p2c - p2n) * (1.0f / KFERMI)));

        float lp = __builtin_nontemporal_load(lp_in + g);
        int tv = s_type[(s + 2) * TSTR + (t + 2)];
        float outv = (lp <= w) ? (float)bn : (float)tv;
        __builtin_nontemporal_store(outv, out_type + g);
    }
}

extern "C" void kernel_launch(void* const* d_in, const int* in_sizes, int n_in,
                              void* d_out, int out_size, void* d_ws, size_t ws_size,
                              hipStream_t stream) {
    (void)in_sizes; (void)n_in; (void)out_size; (void)d_ws; (void)ws_size;
    const int*   type_in = (const int*)  d_in[0];
    const float* Q_in    = (const float*)d_in[1];
    const float* tie_in  = (const float*)d_in[2];
    const int*   rtype   = (const int*)  d_in[3];
    const float* eps_in  = (const float*)d_in[4];
    const int*   dir_in  = (const int*)  d_in[5];
    const float* lp_in   = (const float*)d_in[6];

    const size_t N = (size_t)LSZ * LSZ;
    float* out_type   = (float*)d_out;        // type_t2   [L*L]
    float* out_Q      = out_type + N;         // Q_new     [N*2*2]
    float* out_profit = out_Q + 4 * N;        // profit    [L*L]

    dim3 grid(LSZ / TILE, LSZ / TILE, 1);
    dim3 block(256, 1, 1);   // 8 wave32 waves
    hipLaunchKernelGGL(spgg_fused_kernel, grid, block, 0, stream,
                       type_in, Q_in, tie_in, rtype, eps_in, dir_in, lp_in,
                       out_type, out_Q, out_profit);
}